// SpectralGNN_91173565759559
// MI455X (gfx1250) — compile-verified
//
#include <hip/hip_runtime.h>
#include <hip/hip_bf16.h>

// ---------------------------------------------------------------------------
// SpectralGNN (ChebConv K=3, two layers, mean-pool, FC) for gfx1250 (MI455X).
// Edge scatter dominates bandwidth (~10 GB); GEMMs run on fp32 WMMA
// (v_wmma_f32_16x16x4_f32), one wave = 16x64 output strip, branch-free K-loop
// (NN % 16 == 0 baked in), W repacked so B fragments are contiguous b128s.
// ---------------------------------------------------------------------------

#define NN   100000
#define NE   3200000
#define NG   256
#define IND  128
#define HIDD 64

static_assert(NN % 16 == 0, "row tiles must be exact");

typedef __attribute__((ext_vector_type(2))) float v2f;
typedef __attribute__((ext_vector_type(8))) float v8f;

// ---------------- degree / normalization -----------------------------------

__global__ void degree_kernel(const int* __restrict__ src, float* __restrict__ deg, int ne) {
    int e = blockIdx.x * blockDim.x + threadIdx.x;
    if (e < ne) atomicAdd(&deg[src[e]], 1.0f);
}

__global__ void dinv_kernel(float* __restrict__ deg, int n) {
    int i = blockIdx.x * blockDim.x + threadIdx.x;
    if (i < n) {
        float d = deg[i];
        deg[i] = (d > 0.0f) ? rsqrtf(fmaxf(d, 1.0f)) : 0.0f;
    }
}

__global__ void edgew_kernel(const int* __restrict__ src, const int* __restrict__ dst,
                             const float* __restrict__ dinv, float* __restrict__ ew, int ne) {
    int e = blockIdx.x * blockDim.x + threadIdx.x;
    if (e < ne) ew[e] = -dinv[src[e]] * dinv[dst[e]];
}

// ---------------- scaled-Laplacian scatter (dominant kernel) ----------------
// One edge row handled by D/4 lanes; each lane: float4 gather + 4 f32 atomics.

template <int D>
__global__ void lhat_kernel(const float* __restrict__ v, const int* __restrict__ src,
                            const int* __restrict__ dst, const float* __restrict__ ew,
                            float* __restrict__ out, long long ne) {
    constexpr int TPE = D / 4;
    long long tid = (long long)blockIdx.x * blockDim.x + threadIdx.x;
    long long e   = tid / TPE;
    int dg        = (int)(tid % TPE);
    if (e >= ne) return;
    int s = src[e];
    int d = dst[e];
    float w = ew[e];
    const float4 xv = *(const float4*)(v + (size_t)s * D + dg * 4);
    float* o = out + (size_t)d * D + dg * 4;
    atomicAdd(o + 0, w * xv.x);
    atomicAdd(o + 1, w * xv.y);
    atomicAdd(o + 2, w * xv.z);
    atomicAdd(o + 3, w * xv.w);
}

// Tx2 = 2*Lhat(Tx1) - Tx0, in place on y (y already holds Lhat(Tx1))
__global__ void axpby_kernel(float* __restrict__ y, const float* __restrict__ x0, long long n) {
    long long i = (long long)blockIdx.x * blockDim.x + threadIdx.x;
    if (i < n) y[i] = 2.0f * y[i] - x0[i];
}

// ---------------- W repack for WMMA B fragments ------------------------------
// Wpk[((t*(IN_D/4) + k4)*32 + lane)*8 + ct*2 + p] = W[t][k4*4 + khalf + p][ct*16 + nlo]
// where khalf = (lane>>4)*2, nlo = lane&15.  One lane's B data for all 4 col
// tiles of one K-step becomes 32 contiguous bytes (two b128 loads).

template <int IN_D>
__global__ void pack_w_kernel(const float* __restrict__ W, float* __restrict__ Wpk) {
    constexpr int K4 = IN_D / 4;
    int tid = blockIdx.x * blockDim.x + threadIdx.x;
    if (tid >= 3 * K4 * 32) return;
    int lane = tid & 31;
    int k4   = (tid >> 5) % K4;
    int t    = (tid >> 5) / K4;
    int khalf = (lane >> 4) * 2;
    int nlo   = lane & 15;
    const float* Wm = W + (size_t)t * IN_D * HIDD;
    float* o = Wpk + (size_t)tid * 8;
#pragma unroll
    for (int ct = 0; ct < 4; ++ct) {
        int n = ct * 16 + nlo;
        o[ct * 2 + 0] = Wm[(size_t)(k4 * 4 + khalf + 0) * HIDD + n];
        o[ct * 2 + 1] = Wm[(size_t)(k4 * 4 + khalf + 1) * HIDD + n];
    }
}

// ---------------- ChebConv GEMM: H = relu(Tx0 W0 + Tx1 W1 + Tx2 W2 + b) -----
// One wave => 16 rows x 64 cols (4 accumulators), fp32 WMMA, K stepped by 4.
// No row tail (NN % 16 == 0) => branch-free inner loop, full EXEC everywhere.

template <int IN_D>
__global__ void __launch_bounds__(256) cheb_gemm_kernel(
    const float* __restrict__ Tx0, const float* __restrict__ Tx1,
    const float* __restrict__ Tx2,
    const float* __restrict__ Wpk,   // [3][IN_D/4][32][8] packed
    const float* __restrict__ bias,  // [64]
    float* __restrict__ out) {       // [NN, 64]
    constexpr int K4 = IN_D / 4;
    constexpr int ROW_TILES = NN / 16;

    const int lane = threadIdx.x & 31;
    const int wave = threadIdx.x >> 5;
    const int rt = blockIdx.x * (blockDim.x >> 5) + wave;
    if (rt >= ROW_TILES) return;  // wave-uniform: EXEC all-1s past here

    const int m     = lane & 15;
    const int khalf = (lane >> 4) * 2;  // 0 or 2
    const int nlo   = lane & 15;
    const float* arow0 = Tx0 + (size_t)(rt * 16 + m) * IN_D + khalf;
    const float* arow1 = Tx1 + (size_t)(rt * 16 + m) * IN_D + khalf;
    const float* arow2 = Tx2 + (size_t)(rt * 16 + m) * IN_D + khalf;
    const float* arows[3] = {arow0, arow1, arow2};

    v8f acc0 = {}, acc1 = {}, acc2 = {}, acc3 = {};
#pragma unroll
    for (int t = 0; t < 3; ++t) {
        const float* A  = arows[t];
        const float* Wt = Wpk + ((size_t)t * K4 * 32 + lane) * 8;
        for (int k4 = 0; k4 < K4; ++k4) {
            v2f a = *(const v2f*)(A + k4 * 4);
            const float4 w0 = *(const float4*)(Wt + (size_t)k4 * 32 * 8);      // ct 0,1
            const float4 w1 = *(const float4*)(Wt + (size_t)k4 * 32 * 8 + 4);  // ct 2,3
            v2f b0; b0.x = w0.x; b0.y = w0.y;
            v2f b1; b1.x = w0.z; b1.y = w0.w;
            v2f b2; b2.x = w1.x; b2.y = w1.y;
            v2f b3; b3.x = w1.z; b3.y = w1.w;
            acc0 = __builtin_amdgcn_wmma_f32_16x16x4_f32(false, a, false, b0, (short)0, acc0, false, false);
            acc1 = __builtin_amdgcn_wmma_f32_16x16x4_f32(false, a, false, b1, (short)0, acc1, false, false);
            acc2 = __builtin_amdgcn_wmma_f32_16x16x4_f32(false, a, false, b2, (short)0, acc2, false, false);
            acc3 = __builtin_amdgcn_wmma_f32_16x16x4_f32(false, a, false, b3, (short)0, acc3, false, false);
        }
    }

    // D layout: VGPR r -> row = rt*16 + (lane>>4)*8 + r, col = ct*16 + nlo
    const float bn0 = bias[0 * 16 + nlo];
    const float bn1 = bias[1 * 16 + nlo];
    const float bn2 = bias[2 * 16 + nlo];
    const float bn3 = bias[3 * 16 + nlo];
    const int rbase = rt * 16 + (lane >> 4) * 8;
#pragma unroll
    for (int r = 0; r < 8; ++r) {
        float* orow = out + (size_t)(rbase + r) * HIDD;
        float v0 = acc0[r] + bn0;
        float v1 = acc1[r] + bn1;
        float v2 = acc2[r] + bn2;
        float v3 = acc3[r] + bn3;
        orow[0 * 16 + nlo] = v0 > 0.0f ? v0 : 0.0f;
        orow[1 * 16 + nlo] = v1 > 0.0f ? v1 : 0.0f;
        orow[2 * 16 + nlo] = v2 > 0.0f ? v2 : 0.0f;
        orow[3 * 16 + nlo] = v3 > 0.0f ? v3 : 0.0f;
    }
}

// ---------------- pooling + FC ----------------------------------------------

__global__ void pool_accum_kernel(const float* __restrict__ H, const int* __restrict__ batch,
                                  float* __restrict__ sums, float* __restrict__ counts, int n) {
    int i = blockIdx.x * blockDim.x + threadIdx.x;  // over n*64
    if (i >= n * HIDD) return;
    int node = i >> 6;
    int d    = i & 63;
    int g    = batch[node];
    atomicAdd(&sums[g * HIDD + d], H[i]);
    if (d == 0) atomicAdd(&counts[g], 1.0f);
}

__global__ void fc_kernel(const float* __restrict__ sums, const float* __restrict__ counts,
                          const float* __restrict__ Wfc, const float* __restrict__ bfc,
                          float* __restrict__ out) {
    int g = blockIdx.x * blockDim.x + threadIdx.x;
    if (g >= NG) return;
    float c = fmaxf(counts[g], 1.0f);
    float acc = 0.0f;
#pragma unroll
    for (int d = 0; d < HIDD; ++d) acc += (sums[g * HIDD + d] / c) * Wfc[d];
    out[g] = acc + bfc[0];
}

// ---------------- host-side orchestration ----------------------------------

extern "C" void kernel_launch(void* const* d_in, const int* in_sizes, int n_in,
                              void* d_out, int out_size, void* d_ws, size_t ws_size,
                              hipStream_t stream) {
    (void)in_sizes; (void)n_in; (void)out_size; (void)ws_size;

    const float* x    = (const float*)d_in[0];           // [NN,128]
    const int*   eidx = (const int*)d_in[1];             // [2,NE]
    const int*   batch= (const int*)d_in[2];             // [NN]
    const float* W1   = (const float*)d_in[3];           // [3,128,64]
    const float* b1   = (const float*)d_in[4];           // [64]
    const float* W2   = (const float*)d_in[5];           // [3,64,64]
    const float* b2   = (const float*)d_in[6];           // [64]
    const float* Wfc  = (const float*)d_in[7];           // [64]
    const float* bfc  = (const float*)d_in[8];           // [1]
    float* out = (float*)d_out;                          // [256]

    const int* src = eidx;
    const int* dst = eidx + NE;

    // workspace layout (256B aligned)
    char* ws = (char*)d_ws;
    size_t off = 0;
    auto carve = [&](size_t bytes) -> char* {
        char* p = ws + off;
        off = (off + bytes + 255) & ~(size_t)255;
        return p;
    };
    float* dinv  = (float*)carve(sizeof(float) * NN);          // deg -> dinv in place
    float* ew    = (float*)carve(sizeof(float) * NE);
    float* TxA1  = (float*)carve(sizeof(float) * (size_t)NN * IND);
    float* TxA2  = (float*)carve(sizeof(float) * (size_t)NN * IND);
    float* H1    = (float*)carve(sizeof(float) * (size_t)NN * HIDD);
    float* H2    = (float*)carve(sizeof(float) * (size_t)NN * HIDD);
    float* Wpk1  = (float*)carve(sizeof(float) * 3 * (IND / 4) * 32 * 8);
    float* Wpk2  = (float*)carve(sizeof(float) * 3 * (HIDD / 4) * 32 * 8);
    float* psum  = (float*)carve(sizeof(float) * NG * HIDD);
    float* pcnt  = (float*)carve(sizeof(float) * NG);
    // layer-2 Tx buffers alias layer-1's (dead after layer-1 GEMM)
    float* TxB1 = TxA1;
    float* TxB2 = TxA2;

    const int T = 256;
    auto blocks = [](long long n, int t) { return (unsigned)((n + t - 1) / t); };

    // ---- normalization weights + weight repack ----
    hipMemsetAsync(dinv, 0, sizeof(float) * NN, stream);
    degree_kernel<<<blocks(NE, T), T, 0, stream>>>(src, dinv, NE);
    dinv_kernel<<<blocks(NN, T), T, 0, stream>>>(dinv, NN);
    edgew_kernel<<<blocks(NE, T), T, 0, stream>>>(src, dst, dinv, ew, NE);
    pack_w_kernel<IND><<<blocks(3 * (IND / 4) * 32, T), T, 0, stream>>>(W1, Wpk1);
    pack_w_kernel<HIDD><<<blocks(3 * (HIDD / 4) * 32, T), T, 0, stream>>>(W2, Wpk2);

    // ---- layer 1 (D = 128) ----
    hipMemsetAsync(TxA1, 0, sizeof(float) * (size_t)NN * IND, stream);
    lhat_kernel<IND><<<blocks((long long)NE * (IND / 4), T), T, 0, stream>>>(
        x, src, dst, ew, TxA1, NE);
    hipMemsetAsync(TxA2, 0, sizeof(float) * (size_t)NN * IND, stream);
    lhat_kernel<IND><<<blocks((long long)NE * (IND / 4), T), T, 0, stream>>>(
        TxA1, src, dst, ew, TxA2, NE);
    axpby_kernel<<<blocks((long long)NN * IND, T), T, 0, stream>>>(
        TxA2, x, (long long)NN * IND);
    {
        int row_tiles = NN / 16;
        int waves_per_block = T / 32;
        cheb_gemm_kernel<IND><<<blocks(row_tiles, waves_per_block), T, 0, stream>>>(
            x, TxA1, TxA2, Wpk1, b1, H1);
    }

    // ---- layer 2 (D = 64) ----
    hipMemsetAsync(TxB1, 0, sizeof(float) * (size_t)NN * HIDD, stream);
    lhat_kernel<HIDD><<<blocks((long long)NE * (HIDD / 4), T), T, 0, stream>>>(
        H1, src, dst, ew, TxB1, NE);
    hipMemsetAsync(TxB2, 0, sizeof(float) * (size_t)NN * HIDD, stream);
    lhat_kernel<HIDD><<<blocks((long long)NE * (HIDD / 4), T), T, 0, stream>>>(
        TxB1, src, dst, ew, TxB2, NE);
    axpby_kernel<<<blocks((long long)NN * HIDD, T), T, 0, stream>>>(
        TxB2, H1, (long long)NN * HIDD);
    {
        int row_tiles = NN / 16;
        int waves_per_block = T / 32;
        cheb_gemm_kernel<HIDD><<<blocks(row_tiles, waves_per_block), T, 0, stream>>>(
            H1, TxB1, TxB2, Wpk2, b2, H2);
    }

    // ---- global mean pool + FC ----
    hipMemsetAsync(psum, 0, sizeof(float) * NG * HIDD, stream);
    hipMemsetAsync(pcnt, 0, sizeof(float) * NG, stream);
    pool_accum_kernel<<<blocks((long long)NN * HIDD, T), T, 0, stream>>>(
        H2, batch, psum, pcnt, NN);
    fc_kernel<<<1, NG, 0, stream>>>(psum, pcnt, Wfc, bfc, out);
}